// Block_1563368095940
// MI455X (gfx1250) — compile-verified
//
#include <hip/hip_runtime.h>

// ---------------------------------------------------------------------------
// CDNA5 (gfx1250) transformer block: LN -> QKV GEMM -> gather+RoPE ->
// windowed attention (K=48) -> unpad+proj GEMM -> segment-mean residual ->
// LN -> MLP (fc1+GELU, fc2+residual). All GEMM-shaped math uses
// V_WMMA_F32_16X16X4_F32 (fp32 A/B/C, matches fp32 reference numerics).
// GEMM staging is double-buffered via GLOBAL_LOAD_ASYNC_TO_LDS_B128
// (ASYNCcnt-tracked) when the toolchain declares the builtin.
// ---------------------------------------------------------------------------

typedef float v2f __attribute__((ext_vector_type(2)));
typedef float v8f __attribute__((ext_vector_type(8)));
typedef int v4i __attribute__((vector_size(16)));

#if defined(__has_builtin)
#if __has_builtin(__builtin_amdgcn_global_load_async_to_lds_b128)
#define HAVE_ASYNC_LDS 1
#endif
#if __has_builtin(__builtin_amdgcn_s_wait_asynccnt)
#define HAVE_WAIT_ASYNC 1
#endif
#endif

#if defined(HAVE_ASYNC_LDS)
typedef __attribute__((address_space(1))) v4i* as1_v4i_ptr;
typedef __attribute__((address_space(3))) v4i* as3_v4i_ptr;
#endif

static __device__ __forceinline__ void wait_async_copies() {
#if defined(HAVE_ASYNC_LDS)
#if defined(HAVE_WAIT_ASYNC)
  __builtin_amdgcn_s_wait_asynccnt(0);
#else
  asm volatile("s_wait_asynccnt 0x0" ::: "memory");
#endif
#endif
}

static __device__ __forceinline__ v8f wmma_f32_16x16x4(v2f a, v2f b, v8f c) {
  // 8 args: (neg_a, A, neg_b, B, c_mod, C, reuse_a, reuse_b)
  return __builtin_amdgcn_wmma_f32_16x16x4_f32(false, a, false, b, (short)0, c,
                                               false, false);
}

static __device__ __forceinline__ float gelu_exact(float x) {
  return 0.5f * x * (1.0f + erff(x * 0.7071067811865475f));
}

// ---------------------------------------------------------------------------
// LayerNorm over 128 channels: one wave per row, 4 channels per lane.
// ---------------------------------------------------------------------------
__global__ __launch_bounds__(128) void ln_kernel(const float* __restrict__ X,
                                                 const float* __restrict__ g,
                                                 const float* __restrict__ b,
                                                 float* __restrict__ Y,
                                                 int rows) {
  int lane = threadIdx.x & 31;
  int row = blockIdx.x * 4 + (threadIdx.x >> 5);
  if (row >= rows) return;
  const float4 v = *(const float4*)(X + (long)row * 128 + lane * 4);
  float s = v.x + v.y + v.z + v.w;
#pragma unroll
  for (int mk = 16; mk; mk >>= 1) s += __shfl_xor(s, mk, 32);
  float mean = s * (1.0f / 128.0f);
  float d0 = v.x - mean, d1 = v.y - mean, d2 = v.z - mean, d3 = v.w - mean;
  float q = d0 * d0 + d1 * d1 + d2 * d2 + d3 * d3;
#pragma unroll
  for (int mk = 16; mk; mk >>= 1) q += __shfl_xor(q, mk, 32);
  float rstd = rsqrtf(q * (1.0f / 128.0f) + 1e-5f);
  const float4 gv = *(const float4*)(g + lane * 4);
  const float4 bv = *(const float4*)(b + lane * 4);
  float4 o;
  o.x = d0 * rstd * gv.x + bv.x;
  o.y = d1 * rstd * gv.y + bv.y;
  o.z = d2 * rstd * gv.z + bv.z;
  o.w = d3 * rstd * gv.w + bv.w;
  *(float4*)(Y + (long)row * 128 + lane * 4) = o;
}

// ---------------------------------------------------------------------------
// Stage one 64x32 A tile (with optional row gather) and 64x32 B tile into
// LDS. Async path issues GLOBAL_LOAD_ASYNC_TO_LDS_B128 per lane (16B each);
// sync fallback does load->VGPR->ds_store. OOB A rows: async leaves stale
// LDS (results masked in epilogue), sync zero-fills.
// ---------------------------------------------------------------------------
static __device__ __forceinline__ void gemm_stage(
    const float* __restrict__ A, const int* __restrict__ rowIdx,
    const float* __restrict__ W, int M, int K, int bm, int bn, int tid, int kb,
    float (*as)[33], float (*bs)[33]) {
#pragma unroll
  for (int j = 0; j < 4; ++j) {
    int f = tid + j * 128;  // float4 slot 0..511
    int r = f >> 3;         // tile row 0..63
    int cv = (f & 7) << 2;  // col 0,4,...,28
    int gr = bm * 64 + r;
    long gn = (long)bn * 64 + r;  // always < Nc (Nc % 64 == 0)
#if defined(HAVE_ASYNC_LDS)
    if (gr < M) {
      long ar = rowIdx ? (long)rowIdx[gr] : (long)gr;
      __builtin_amdgcn_global_load_async_to_lds_b128(
          (as1_v4i_ptr)(A + ar * K + kb + cv), (as3_v4i_ptr)&as[r][cv], 0, 0);
    }
    __builtin_amdgcn_global_load_async_to_lds_b128(
        (as1_v4i_ptr)(W + gn * K + kb + cv), (as3_v4i_ptr)&bs[r][cv], 0, 0);
#else
    float4 av = make_float4(0.f, 0.f, 0.f, 0.f);
    if (gr < M) {
      long ar = rowIdx ? (long)rowIdx[gr] : (long)gr;
      av = *(const float4*)(A + ar * K + kb + cv);
    }
    as[r][cv + 0] = av.x;
    as[r][cv + 1] = av.y;
    as[r][cv + 2] = av.z;
    as[r][cv + 3] = av.w;
    float4 bv4 = *(const float4*)(W + gn * K + kb + cv);
    bs[r][cv + 0] = bv4.x;
    bs[r][cv + 1] = bv4.y;
    bs[r][cv + 2] = bv4.z;
    bs[r][cv + 3] = bv4.w;
#endif
  }
}

// ---------------------------------------------------------------------------
// Generic WMMA-f32 GEMM: C[M,Nc] = epi(A'[M,K] @ W[Nc,K]^T + bias)
//   A' = rowIdx ? A[rowIdx[m]] : A[m]   (fused gather for proj/unpad)
//   epi: 0=none, 1=exact GELU, 2=add res[m,n] (residual; res may alias Cout)
// Block = 128 threads (4 waves), tile 64x64, each wave does 32x32 (2x2 WMMA).
// Double-buffered LDS: one barrier per K-chunk, copies overlap compute.
// Nc must be a multiple of 64; K a multiple of 32; M arbitrary.
// ---------------------------------------------------------------------------
__global__ __launch_bounds__(128) void gemm_wmma_kernel(
    const float* __restrict__ A, const int* __restrict__ rowIdx,
    const float* __restrict__ W, const float* __restrict__ bias, float* Cout,
    const float* res, int M, int Nc, int K, int epi) {
  __shared__ float As[2][64][33];
  __shared__ float Bs[2][64][33];
  const int tid = threadIdx.x;
  const int lane = tid & 31;
  const int wv = tid >> 5;
  const int bm = blockIdx.y, bn = blockIdx.x;
  const int rowBase = (wv >> 1) * 32;
  const int colBase = (wv & 1) * 32;
  const int m = lane & 15;
  const int kh = (lane >> 4) * 2;

  v8f acc[2][2];
#pragma unroll
  for (int i = 0; i < 2; ++i)
#pragma unroll
    for (int j = 0; j < 2; ++j) acc[i][j] = (v8f)(0.0f);

  const int nch = K >> 5;
  gemm_stage(A, rowIdx, W, M, K, bm, bn, tid, 0, As[0], Bs[0]);
  wait_async_copies();
  __syncthreads();

  for (int t = 0; t < nch; ++t) {
    const int cur = t & 1;
    if (t + 1 < nch)  // issue next chunk's copies before compute (overlap)
      gemm_stage(A, rowIdx, W, M, K, bm, bn, tid, (t + 1) * 32, As[cur ^ 1],
                 Bs[cur ^ 1]);
    const float(*pa)[33] = As[cur];
    const float(*pb)[33] = Bs[cur];
#pragma unroll
    for (int kk = 0; kk < 32; kk += 4) {
      v2f a[2], bfrag[2];
#pragma unroll
      for (int ri = 0; ri < 2; ++ri) {
        a[ri].x = pa[rowBase + ri * 16 + m][kk + kh + 0];
        a[ri].y = pa[rowBase + ri * 16 + m][kk + kh + 1];
      }
#pragma unroll
      for (int ci = 0; ci < 2; ++ci) {
        bfrag[ci].x = pb[colBase + ci * 16 + m][kk + kh + 0];
        bfrag[ci].y = pb[colBase + ci * 16 + m][kk + kh + 1];
      }
#pragma unroll
      for (int ri = 0; ri < 2; ++ri)
#pragma unroll
        for (int ci = 0; ci < 2; ++ci)
          acc[ri][ci] = wmma_f32_16x16x4(a[ri], bfrag[ci], acc[ri][ci]);
    }
    wait_async_copies();
    __syncthreads();
  }

  // Epilogue: C/D layout -> element (row = v + 8*(lane>>4), col = lane&15)
#pragma unroll
  for (int ri = 0; ri < 2; ++ri)
#pragma unroll
    for (int ci = 0; ci < 2; ++ci) {
      int gcol = bn * 64 + colBase + ci * 16 + (lane & 15);
      float bval = bias ? bias[gcol] : 0.0f;
#pragma unroll
      for (int v = 0; v < 8; ++v) {
        int grow = bm * 64 + rowBase + ri * 16 + v + 8 * (lane >> 4);
        if (grow < M) {
          float x = acc[ri][ci][v] + bval;
          if (epi == 1)
            x = gelu_exact(x);
          else if (epi == 2)
            x += res[(long)grow * Nc + gcol];
          Cout[(long)grow * Nc + gcol] = x;
        }
      }
    }
}

// ---------------------------------------------------------------------------
// Gather qkv rows via path_inverse[pad[i]] and apply RoPE to q,k.
// One block (128 threads) per padded row; thread = channel.
// ---------------------------------------------------------------------------
__global__ __launch_bounds__(128) void gather_rope_kernel(
    const float* __restrict__ qkv, const int* __restrict__ pinv,
    const int* __restrict__ pad, const float* __restrict__ cosb,
    const float* __restrict__ sinb, float* __restrict__ qp,
    float* __restrict__ kp, float* __restrict__ vp) {
  int i = blockIdx.x;
  int c = threadIdx.x;
  int pr = pad[i];
  int src = pinv[pr];
  long qb = (long)src * 384;
  int d = c & 15;
  float qo, ko;
  if (d < 8) {
    float cv = cosb[(long)pr * 16 + d];
    float sv = sinb[(long)pr * 16 + d];
    float q1 = qkv[qb + c], q2 = qkv[qb + c + 8];
    qo = q1 * cv - q2 * sv;
    float k1 = qkv[qb + 128 + c], k2 = qkv[qb + 128 + c + 8];
    ko = k1 * cv - k2 * sv;
  } else {
    float cv = cosb[(long)pr * 16 + d - 8];
    float sv = sinb[(long)pr * 16 + d - 8];
    float q2 = qkv[qb + c], q1 = qkv[qb + c - 8];
    qo = q2 * cv + q1 * sv;
    float k2 = qkv[qb + 128 + c], k1 = qkv[qb + 128 + c - 8];
    ko = k2 * cv + k1 * sv;
  }
  float vo = qkv[qb + 256 + c];
  long o = (long)i * 128 + c;
  qp[o] = qo;
  kp[o] = ko;
  vp[o] = vo;
}

// ---------------------------------------------------------------------------
// Windowed attention: one wave per (window, head). 48x48 scores via 3x3 WMMA
// tiles (K=16 in 4 steps), register softmax (width-16 shuffles), P staged in
// LDS to convert C-layout -> A-layout, then P@V via 3 WMMA tiles x 12 k-steps.
// ---------------------------------------------------------------------------
__global__ __launch_bounds__(128) void attn_kernel(const float* __restrict__ qp,
                                                   const float* __restrict__ kp,
                                                   const float* __restrict__ vp,
                                                   float* __restrict__ op) {
  __shared__ float p_s[4][48][49];
  const int lane = threadIdx.x & 31;
  const int wv = threadIdx.x >> 5;
  const int pair = blockIdx.x * 4 + wv;
  const int w = pair >> 3;
  const int hh = pair & 7;
  const long base = (long)w * 48;
  const float* qb = qp + base * 128 + hh * 16;
  const float* kb = kp + base * 128 + hh * 16;
  const float* vb = vp + base * 128 + hh * 16;
  const int m = lane & 15;
  const int kh = (lane >> 4) * 2;

  v8f S[3][3];
#pragma unroll
  for (int i = 0; i < 3; ++i)
#pragma unroll
    for (int j = 0; j < 3; ++j) S[i][j] = (v8f)(0.0f);

  // scores = q @ k^T  (M=48, N=48, K=16)
#pragma unroll
  for (int s = 0; s < 4; ++s) {
    v2f a[3], b[3];
#pragma unroll
    for (int i = 0; i < 3; ++i) {
      const float* pa = qb + (long)(i * 16 + m) * 128 + s * 4 + kh;
      a[i].x = pa[0];
      a[i].y = pa[1];
      const float* pb = kb + (long)(i * 16 + m) * 128 + s * 4 + kh;
      b[i].x = pb[0];
      b[i].y = pb[1];
    }
#pragma unroll
    for (int i = 0; i < 3; ++i)
#pragma unroll
      for (int j = 0; j < 3; ++j) S[i][j] = wmma_f32_16x16x4(a[i], b[j], S[i][j]);
  }

  // softmax over 48 columns per row (scale = D^-0.5 = 0.25)
#pragma unroll
  for (int i = 0; i < 3; ++i)
#pragma unroll
    for (int v = 0; v < 8; ++v) {
      float x0 = S[i][0][v] * 0.25f;
      float x1 = S[i][1][v] * 0.25f;
      float x2 = S[i][2][v] * 0.25f;
      float mx = fmaxf(fmaxf(x0, x1), x2);
#pragma unroll
      for (int mk = 8; mk; mk >>= 1) mx = fmaxf(mx, __shfl_xor(mx, mk, 16));
      float e0 = __expf(x0 - mx);
      float e1 = __expf(x1 - mx);
      float e2 = __expf(x2 - mx);
      float sm = e0 + e1 + e2;
#pragma unroll
      for (int mk = 8; mk; mk >>= 1) sm += __shfl_xor(sm, mk, 16);
      float inv = 1.0f / sm;
      S[i][0][v] = e0 * inv;
      S[i][1][v] = e1 * inv;
      S[i][2][v] = e2 * inv;
    }

  // Stage P into LDS (C-layout -> row-major) for A-fragment reads.
#pragma unroll
  for (int i = 0; i < 3; ++i)
#pragma unroll
    for (int j = 0; j < 3; ++j)
#pragma unroll
      for (int v = 0; v < 8; ++v)
        p_s[wv][i * 16 + v + 8 * (lane >> 4)][j * 16 + (lane & 15)] = S[i][j][v];
  __syncthreads();

  // o = P @ V  (M=48, N=16, K=48)
#pragma unroll
  for (int i = 0; i < 3; ++i) {
    v8f acc = (v8f)(0.0f);
#pragma unroll
    for (int s = 0; s < 12; ++s) {
      v2f a, b;
      a.x = p_s[wv][i * 16 + m][s * 4 + kh + 0];
      a.y = p_s[wv][i * 16 + m][s * 4 + kh + 1];
      int kr = s * 4 + kh;
      b.x = vb[(long)(kr + 0) * 128 + (lane & 15)];
      b.y = vb[(long)(kr + 1) * 128 + (lane & 15)];
      acc = wmma_f32_16x16x4(a, b, acc);
    }
#pragma unroll
    for (int v = 0; v < 8; ++v) {
      int row = i * 16 + v + 8 * (lane >> 4);
      op[(base + row) * 128 + hh * 16 + (lane & 15)] = acc[v];
    }
  }
}

// ---------------------------------------------------------------------------
// Scatter-add projected rows into per-point sums + counts (segment_sum).
// ---------------------------------------------------------------------------
__global__ void scatter_kernel(const float* __restrict__ P,
                               const int* __restrict__ pinv,
                               float* __restrict__ sums,
                               float* __restrict__ cnt, long total) {
  long t = (long)blockIdx.x * 256 + threadIdx.x;
  if (t >= total) return;
  int i = (int)(t >> 7);
  int c = (int)(t & 127);
  int dst = pinv[i];
  atomicAdd(&sums[(long)dst * 128 + c], P[t]);
  if (c == 0) atomicAdd(&cnt[dst], 1.0f);
}

// ---------------------------------------------------------------------------
// x = feat + sums / max(cnt, 1)
// ---------------------------------------------------------------------------
__global__ void xres_kernel(const float* __restrict__ feat,
                            const float* __restrict__ sums,
                            const float* __restrict__ cnt,
                            float* __restrict__ out, long total) {
  long t = (long)blockIdx.x * 256 + threadIdx.x;
  if (t >= total) return;
  int r = (int)(t >> 7);
  out[t] = feat[t] + sums[t] / fmaxf(cnt[r], 1.0f);
}

// ---------------------------------------------------------------------------
extern "C" void kernel_launch(void* const* d_in, const int* in_sizes, int n_in,
                              void* d_out, int out_size, void* d_ws,
                              size_t ws_size, hipStream_t stream) {
  const float* feat = (const float*)d_in[0];
  const float* cosb = (const float*)d_in[1];
  const float* sinb = (const float*)d_in[2];
  const float* g1 = (const float*)d_in[3];
  const float* b1 = (const float*)d_in[4];
  const float* qkv_w = (const float*)d_in[5];
  const float* qkv_b = (const float*)d_in[6];
  const float* proj_w = (const float*)d_in[7];
  const float* proj_b = (const float*)d_in[8];
  const float* g2 = (const float*)d_in[9];
  const float* b2 = (const float*)d_in[10];
  const float* fc1_w = (const float*)d_in[11];
  const float* fc1_b = (const float*)d_in[12];
  const float* fc2_w = (const float*)d_in[13];
  const float* fc2_b = (const float*)d_in[14];
  const int* pinv = (const int*)d_in[15];
  const int* pad = (const int*)d_in[16];
  const int* unpad = (const int*)d_in[17];

  const int N = in_sizes[0] / 128;
  const int Mpad = in_sizes[16];
  const int M = in_sizes[17];
  const int Wn = Mpad / 48;

  // Workspace layout (floats), with buffer reuse across pipeline stages.
  float* ws = (float*)d_ws;
  size_t off = 0;
  float* h = ws + off;  // LN1 out; later: sums
  off += (size_t)N * 128;
  size_t r1 = (size_t)N * 384;
  if ((size_t)Mpad * 128 > r1) r1 = (size_t)Mpad * 128;
  float* qkvbuf = ws + off;  // qkv; later: o_pad
  off += r1;
  float* qp = ws + off;  // padded q; later: proj out
  off += (size_t)Mpad * 128;
  float* kp = ws + off;  // padded k; later: h2 (LN2 out)
  off += (size_t)Mpad * 128;
  float* vp = ws + off;  // padded v
  off += (size_t)Mpad * 128;
  float* mid = ws + off;  // MLP hidden (N x 512)
  off += (size_t)N * 512;
  float* cnt = ws + off;  // segment counts
  off += (size_t)N;
  float* out = (float*)d_out;

  // 1. h = LN1(feat)
  ln_kernel<<<(N + 3) / 4, 128, 0, stream>>>(feat, g1, b1, h, N);
  // 2. qkv = h @ qkv_w^T + qkv_b
  gemm_wmma_kernel<<<dim3(384 / 64, (N + 63) / 64), 128, 0, stream>>>(
      h, nullptr, qkv_w, qkv_b, qkvbuf, nullptr, N, 384, 128, 0);
  // 3. gather via path_inverse[pad] + RoPE -> qp, kp, vp
  gather_rope_kernel<<<Mpad, 128, 0, stream>>>(qkvbuf, pinv, pad, cosb, sinb,
                                               qp, kp, vp);
  // 4. windowed attention -> o_pad (reuses qkv buffer)
  float* opad = qkvbuf;
  attn_kernel<<<Wn * 2, 128, 0, stream>>>(qp, kp, vp, opad);
  // 5. proj with fused unpad gather -> pout (reuses qp)
  float* pout = qp;
  gemm_wmma_kernel<<<dim3(128 / 64, (M + 63) / 64), 128, 0, stream>>>(
      opad, unpad, proj_w, proj_b, pout, nullptr, M, 128, 128, 0);
  // 6. zero sums (reuses h) and counts
  (void)hipMemsetAsync(h, 0, (size_t)N * 128 * sizeof(float), stream);
  (void)hipMemsetAsync(cnt, 0, (size_t)N * sizeof(float), stream);
  // 7. segment scatter-add
  long totM = (long)M * 128;
  scatter_kernel<<<(unsigned)((totM + 255) / 256), 256, 0, stream>>>(
      pout, pinv, h, cnt, totM);
  // 8. x = feat + mean -> d_out
  long totN = (long)N * 128;
  xres_kernel<<<(unsigned)((totN + 255) / 256), 256, 0, stream>>>(feat, h, cnt,
                                                                  out, totN);
  // 9. h2 = LN2(x) (reuses kp)
  float* h2 = kp;
  ln_kernel<<<(N + 3) / 4, 128, 0, stream>>>(out, g2, b2, h2, N);
  // 10. mid = GELU(h2 @ fc1_w^T + fc1_b)
  gemm_wmma_kernel<<<dim3(512 / 64, (N + 63) / 64), 128, 0, stream>>>(
      h2, nullptr, fc1_w, fc1_b, mid, nullptr, N, 512, 128, 1);
  // 11. out = x + (mid @ fc2_w^T + fc2_b)   (res aliases out; 1 thread/elem)
  gemm_wmma_kernel<<<dim3(128 / 64, (N + 63) / 64), 128, 0, stream>>>(
      mid, nullptr, fc2_w, fc2_b, out, out, N, 128, 512, 2);
}